// AnchorTarget_66254165508671
// MI455X (gfx1250) — compile-verified
//
#include <hip/hip_runtime.h>
#include <math.h>
#include <stdint.h>

// ---------------------------------------------------------------------------
// AnchorTarget (RPN anchor target layer) for MI455X / gfx1250.
//
// d_in[0] = scores   (1,H,W,18)  -- only shape used (H=W=sqrt(size/18))
// d_in[1] = im_info  (1,3)       -- [img_h, img_w, scale]
// d_in[2] = gt_boxes (1,K,4)
// d_out   = labels (N) ++ bbox_targets (N,4), N = H*W*9
//
// Design (see analysis): no matmul structure exists -> no WMMA. CDNA5 paths
// used: async global->LDS staging of the gt table (ASYNCcnt) in phase 1,
// wave32 shuffle reductions + total-order u64 atomicMax argmax in phase 2.
// ---------------------------------------------------------------------------

#define NUM_A 9

// Canonical base anchors (base_size=16, ratios 0.5/1/2, scales 8/16/32),
// exactly integer-valued -> exact in fp32. Layout: x1,y1,x2,y2 per anchor.
__device__ const float c_base[NUM_A * 4] = {
    -84.f,  -40.f,  99.f,  55.f,
   -176.f,  -88.f, 191.f, 103.f,
   -360.f, -184.f, 375.f, 199.f,
    -56.f,  -56.f,  71.f,  71.f,
   -120.f, -120.f, 135.f, 135.f,
   -248.f, -248.f, 263.f, 263.f,
    -36.f,  -80.f,  51.f,  95.f,
    -80.f, -168.f,  95.f, 183.f,
   -168.f, -344.f, 183.f, 359.f,
};

__device__ __forceinline__ void anchor_at(int i, int W,
                                          float& x1, float& y1,
                                          float& x2, float& y2) {
  int a   = i % NUM_A;
  int pos = i / NUM_A;
  float sx = (float)(pos % W) * 8.0f;
  float sy = (float)(pos / W) * 8.0f;
  x1 = sx + c_base[a * 4 + 0];
  y1 = sy + c_base[a * 4 + 1];
  x2 = sx + c_base[a * 4 + 2];
  y2 = sy + c_base[a * 4 + 3];
}

__device__ __forceinline__ float iou_of(float ax1, float ay1, float ax2, float ay2,
                                        float area_a,
                                        float g0, float g1, float g2, float g3,
                                        float area_g) {
  float iw = fminf(ax2, g2) - fmaxf(ax1, g0) + 1.0f;
  float ih = fminf(ay2, g3) - fmaxf(ay1, g1) + 1.0f;
  iw = fmaxf(iw, 0.0f);
  ih = fmaxf(ih, 0.0f);
  float inter = iw * ih;
  return inter / (area_a + area_g - inter);
}

// ---------------------------------------------------------------------------
// Phase 1: one thread per anchor. gt table staged into LDS with CDNA5 async
// global->LDS loads (wave 0 issues, s_wait_asynccnt 0, then block barrier).
// Produces: preliminary labels (without the per-gt-winner rule) + all bbox
// regression targets.
// ---------------------------------------------------------------------------
__global__ __launch_bounds__(256) void at_phase1(
    const float* __restrict__ gt, const float* __restrict__ im_info,
    float* __restrict__ labels, float* __restrict__ targets,
    int N, int K, int W) {
  __shared__ float sgt[128 * 4];  // K <= 128 (harness: K = 128)

  if (threadIdx.x < 32) {
    unsigned lds_base = (unsigned)(uintptr_t)(&sgt[0]);  // low 32 bits = LDS offset
    int nbytes = K * 16;
    const char* gp = (const char*)gt;
    for (int ofs = (int)threadIdx.x * 16; ofs < nbytes; ofs += 512) {
      unsigned laddr = lds_base + (unsigned)ofs;
      const char* gaddr = gp + ofs;
      // CDNA5 async memory->LDS copy, tracked with ASYNCcnt (GV mode).
      asm volatile("global_load_async_to_lds_b128 %0, %1, off"
                   :: "v"(laddr), "v"(gaddr) : "memory");
    }
    asm volatile("s_wait_asynccnt 0x0" ::: "memory");
  }
  __syncthreads();

  int i = blockIdx.x * blockDim.x + threadIdx.x;
  if (i >= N) return;

  float ax1, ay1, ax2, ay2;
  anchor_at(i, W, ax1, ay1, ax2, ay2);
  float area_a = (ax2 - ax1 + 1.0f) * (ay2 - ay1 + 1.0f);

  const float4* sg4 = (const float4*)sgt;
  float best = -1.0f;  // all IoU >= 0 -> k=0 wins ties (first occurrence)
  int bk = 0;
  for (int k = 0; k < K; ++k) {
    float4 g = sg4[k];
    float area_g = (g.z - g.x + 1.0f) * (g.w - g.y + 1.0f);
    float v = iou_of(ax1, ay1, ax2, ay2, area_a, g.x, g.y, g.z, g.w, area_g);
    if (v > best) { best = v; bk = k; }
  }

  // bbox_transform(anchor, gt[argmax])
  {
    float4 g = sg4[bk];
    float ew = ax2 - ax1 + 1.0f;
    float eh = ay2 - ay1 + 1.0f;
    float ecx = ax1 + 0.5f * ew;
    float ecy = ay1 + 0.5f * eh;
    float gw = g.z - g.x + 1.0f;
    float gh = g.w - g.y + 1.0f;
    float gcx = g.x + 0.5f * gw;
    float gcy = g.y + 0.5f * gh;
    float4 t;
    t.x = (gcx - ecx) / ew;
    t.y = (gcy - ecy) / eh;
    t.z = logf(gw / ew);
    t.w = logf(gh / eh);
    ((float4*)targets)[i] = t;
  }

  // Preliminary label (per-gt winners patched in phase 3).
  float imh = im_info[0];
  float imw = im_info[1];
  bool inside = (ax1 >= 0.0f) && (ay1 >= 0.0f) && (ax2 < imw) && (ay2 < imh);
  float lab = -1.0f;
  if (inside) {
    if (best >= 0.7f)      lab = 1.0f;
    else if (best < 0.3f)  lab = 0.0f;
  }
  labels[i] = lab;
}

// ---------------------------------------------------------------------------
// Phase 2: per-gt argmax over all N anchors, chunked across blocks to fill the
// GPU. Total-order key = (iou_bits << 32) | ~anchor_idx reproduces JAX's
// first-occurrence argmax exactly; atomicMax combine is deterministic.
// ---------------------------------------------------------------------------
__global__ __launch_bounds__(256) void at_phase2(
    const float* __restrict__ gt, unsigned long long* __restrict__ keys,
    int N, int W, int nchunks) {
  int k = blockIdx.x;       // gt index
  int chunk = blockIdx.y;

  float g0 = gt[k * 4 + 0], g1 = gt[k * 4 + 1];
  float g2 = gt[k * 4 + 2], g3 = gt[k * 4 + 3];
  float area_g = (g2 - g0 + 1.0f) * (g3 - g1 + 1.0f);

  int per = (N + nchunks - 1) / nchunks;
  int i0 = chunk * per;
  int i1 = min(N, i0 + per);

  unsigned long long bestkey = 0ULL;
  for (int i = i0 + (int)threadIdx.x; i < i1; i += (int)blockDim.x) {
    float ax1, ay1, ax2, ay2;
    anchor_at(i, W, ax1, ay1, ax2, ay2);
    float area_a = (ax2 - ax1 + 1.0f) * (ay2 - ay1 + 1.0f);
    float v = iou_of(ax1, ay1, ax2, ay2, area_a, g0, g1, g2, g3, area_g);
    unsigned long long key =
        ((unsigned long long)__float_as_uint(v) << 32) |
        (unsigned long long)(0xFFFFFFFFu - (unsigned)i);
    if (key > bestkey) bestkey = key;
  }

  // wave32 reduction
  for (int off = 16; off > 0; off >>= 1) {
    unsigned long long o = __shfl_down(bestkey, off, 32);
    if (o > bestkey) bestkey = o;
  }
  __shared__ unsigned long long red[8];
  if ((threadIdx.x & 31u) == 0u) red[threadIdx.x >> 5] = bestkey;
  __syncthreads();
  if (threadIdx.x == 0) {
    unsigned long long b = red[0];
    int nw = (int)(blockDim.x >> 5);
    for (int w = 1; w < nw; ++w)
      if (red[w] > b) b = red[w];
    atomicMax(&keys[k], b);
  }
}

// ---------------------------------------------------------------------------
// Phase 3: decode winners; an anchor that is the argmax of EXACTLY one gt
// (update_mask == 1) and is inside the image gets label 1.
// ---------------------------------------------------------------------------
__global__ void at_phase3(const unsigned long long* __restrict__ keys,
                          const float* __restrict__ im_info,
                          float* __restrict__ labels, int K, int W) {
  int t = (int)threadIdx.x;
  if (t >= K) return;
  unsigned idx = 0xFFFFFFFFu - (unsigned)(keys[t] & 0xFFFFFFFFull);
  int cnt = 0;
  for (int s = 0; s < K; ++s) {
    unsigned o = 0xFFFFFFFFu - (unsigned)(keys[s] & 0xFFFFFFFFull);
    cnt += (o == idx) ? 1 : 0;
  }
  if (cnt == 1) {
    float ax1, ay1, ax2, ay2;
    anchor_at((int)idx, W, ax1, ay1, ax2, ay2);
    float imh = im_info[0];
    float imw = im_info[1];
    bool inside = (ax1 >= 0.0f) && (ay1 >= 0.0f) && (ax2 < imw) && (ay2 < imh);
    if (inside) labels[idx] = 1.0f;
  }
}

// ---------------------------------------------------------------------------
extern "C" void kernel_launch(void* const* d_in, const int* in_sizes, int n_in,
                              void* d_out, int out_size, void* d_ws, size_t ws_size,
                              hipStream_t stream) {
  (void)n_in; (void)out_size; (void)ws_size;
  const float* im_info = (const float*)d_in[1];
  const float* gt      = (const float*)d_in[2];

  int HW = in_sizes[0] / (2 * NUM_A);      // H*W from scores (1,H,W,18)
  int W  = (int)(sqrt((double)HW) + 0.5);  // square feature map
  int H  = HW / W;
  int N  = H * W * NUM_A;
  int K  = in_sizes[2] / 4;

  float* labels  = (float*)d_out;
  float* targets = labels + N;             // (N,4) row-major, 16B-aligned

  unsigned long long* keys = (unsigned long long*)d_ws;  // K u64 keys
  hipMemsetAsync(d_ws, 0, (size_t)K * sizeof(unsigned long long), stream);

  at_phase1<<<(N + 255) / 256, 256, 0, stream>>>(gt, im_info, labels, targets,
                                                 N, K, W);

  const int nchunks = 16;
  at_phase2<<<dim3((unsigned)K, nchunks), 256, 0, stream>>>(gt, keys, N, W,
                                                            nchunks);

  at_phase3<<<1, K, 0, stream>>>(keys, im_info, labels, K, W);
}